// BothConvLayer_37168646980098
// MI455X (gfx1250) — compile-verified
//
#include <hip/hip_runtime.h>

typedef float v2f __attribute__((ext_vector_type(2)));
typedef float v8f __attribute__((ext_vector_type(8)));

#define KIN   64
#define KOUT  64
#define NSP   24
#define NB    64                 // batch
#define TILE  (64 * 64)          // one 64x64 fp32 block (floats)
#define LDSW  68                 // padded LDS row stride (floats): conflict-free b64 reads
#define TILE_LDS_BYTES (64 * LDSW * 4)   // 17408, 16B-aligned rows (272 = 17*16)
#define LDS_TOTAL_BYTES (4 * TILE_LDS_BYTES)  // A0,B0,A1,B1 double buffer

// ---------------------------------------------------------------------------
// Prep: Wd[s][o][k] = W0-W1, W1[s][o][k];  Xm[d][j][b][k] = x[b,j,k,d];
//       Xs[j][b][k] = sum_c x[b,j,k,c]
// ---------------------------------------------------------------------------
__global__ __launch_bounds__(256) void prep_kernel(
    const float* __restrict__ x, const float* __restrict__ w,
    float* __restrict__ Xm, float* __restrict__ Xs,
    float* __restrict__ Wd, float* __restrict__ W1)
{
  int t = blockIdx.x * 256 + threadIdx.x;
  if (t >= NSP * KIN * NB) return;               // 98304 threads per plane
  if (blockIdx.y == 0) {
    // x plane: t = (j*64 + b)*64 + k
    int k = t & 63, b = (t >> 6) & 63, j = t >> 12;
    const float* xp = x + (((size_t)b * 24 + j) * KIN + k) * 6;
    float s = 0.f;
#pragma unroll
    for (int c = 0; c < 6; ++c) {
      float v = xp[c];
      s += v;
      Xm[(((size_t)c * NSP + j) * NB + b) * KIN + k] = v;
    }
    Xs[((size_t)j * NB + b) * KIN + k] = s;
  } else {
    // weight plane: t = (s*64 + o)*64 + k ; input w[o][k][s][c]
    int k = t & 63, o = (t >> 6) & 63, sidx = t >> 12;
    const float* wp = w + (((size_t)o * KIN + k) * NSP + sidx) * 2;
    float w0 = wp[0], w1v = wp[1];
    Wd[t] = w0 - w1v;
    W1[t] = w1v;
  }
}

// ---------------------------------------------------------------------------
// GEMM: per (i[,d]) block computes Out[64 o][64 b] = sum_j A[sp[i,j]] * X[j]
// Tiles are DMA'd into LDS with GLOBAL_LOAD_ASYNC_TO_LDS_B128 (ASYNCcnt),
// double-buffered: issue 8 async b128/wave for block j+1, s_wait_asynccnt 8
// guarantees block j's 8 are complete (per-wave in-order completion).
// mode 0: A=W1, X=Xs -> Gs[i][o][b]
// mode 1: A=Wd, X=Xm[d] -> out[b][i][o*6+d] = acc + Gs[i][o][b] + bias[o]
// ---------------------------------------------------------------------------
__global__ __launch_bounds__(256) void gemm_kernel(
    const float* __restrict__ A,      // [24][64][64]  (s-major blocks)
    const float* __restrict__ X,      // mode1: [6][24][64][64]; mode0: [24][64][64]
    const int*   __restrict__ sp,     // [24][24]
    const float* __restrict__ Gs,     // mode1: read [24][64][64]
    const float* __restrict__ bias,   // mode1: read [64]
    float*       __restrict__ outp,   // mode0: Gs; mode1: out [64][24][384]
    int mode)
{
  extern __shared__ float smem[];
  const unsigned lds_base = (unsigned)__builtin_amdgcn_groupstaticsize(); // == 0 here

  const int i = blockIdx.x;
  const int d = blockIdx.y;                        // 0 in mode 0
  const float* Xb = X + (mode ? (size_t)d * (NSP * NB * KIN) : 0);
  const int* spi = sp + i * 24;

  const int tid  = threadIdx.x;
  const int wave = tid >> 5;
  const int lane = tid & 31;
  const int half = lane >> 4;                      // K-half select for 16x16x4 f32
  const int l15  = lane & 15;
  const int mt   = wave >> 1;                      // M tile 0..3 (o)
  const int nt0  = (wave & 1) << 1;                // N tiles {nt0, nt0+1} (b)
  const int arow  = mt * 16 + l15;
  const int brow0 = nt0 * 16 + l15;
  const int brow1 = brow0 + 16;

  // Per-thread async-copy slots: 4 x b128 per tile. Global tile is dense
  // (idx4*16 bytes); LDS slot is row-padded, still 16B aligned.
  unsigned goff[4], loff[4];
#pragma unroll
  for (int p = 0; p < 4; ++p) {
    int idx4 = tid + p * 256;                      // 1024 b128 per tile
    int row = idx4 >> 4, col = (idx4 & 15) << 2;
    goff[p] = (unsigned)idx4 * 16u;
    loff[p] = lds_base + ((unsigned)(row * LDSW + col)) * 4u;
  }

  auto issue_block = [&](const float* ga, const float* gb, unsigned bufbase) {
#pragma unroll
    for (int p = 0; p < 4; ++p) {
      unsigned la = loff[p] + bufbase;
      unsigned lb = la + (unsigned)TILE_LDS_BYTES;
      asm volatile("global_load_async_to_lds_b128 %0, %1, %2"
                   :: "v"(la), "v"(goff[p]), "s"(ga) : "memory");
      asm volatile("global_load_async_to_lds_b128 %0, %1, %2"
                   :: "v"(lb), "v"(goff[p]), "s"(gb) : "memory");
    }
  };

  v8f acc0 = {0.f, 0.f, 0.f, 0.f, 0.f, 0.f, 0.f, 0.f};
  v8f acc1 = {0.f, 0.f, 0.f, 0.f, 0.f, 0.f, 0.f, 0.f};

  // Prologue: DMA block j=0 into buffer 0.
  issue_block(A + (size_t)spi[0] * TILE, Xb, 0u);

  for (int j = 0; j < 24; ++j) {
    const unsigned cur = (j & 1) ? 2u * TILE_LDS_BYTES : 0u;
    const unsigned nxt = (j & 1) ? 0u : 2u * TILE_LDS_BYTES;

    __syncthreads();                   // all waves done computing on `nxt` buffer
    if (j + 1 < 24) {
      issue_block(A + (size_t)spi[j + 1] * TILE, Xb + (size_t)(j + 1) * TILE, nxt);
      asm volatile("s_wait_asynccnt 0x8" ::: "memory");  // block j's 8 DMAs done
    } else {
      asm volatile("s_wait_asynccnt 0x0" ::: "memory");
    }
    __syncthreads();                   // every wave's DMA for `cur` visible

    const float* la = smem + (cur >> 2);
    const float* lb = la + 64 * LDSW;
#pragma unroll
    for (int kk = 0; kk < 16; ++kk) {  // K=64 per block, 4 per wmma
      int kb = kk * 4 + half * 2;
      v2f av = *(const v2f*)&la[arow  * LDSW + kb];
      v2f b0 = *(const v2f*)&lb[brow0 * LDSW + kb];
      v2f b1 = *(const v2f*)&lb[brow1 * LDSW + kb];
      acc0 = __builtin_amdgcn_wmma_f32_16x16x4_f32(false, av, false, b0,
                                                   (short)0, acc0, false, false);
      acc1 = __builtin_amdgcn_wmma_f32_16x16x4_f32(false, av, false, b1,
                                                   (short)0, acc1, false, false);
    }
  }

  // Epilogue. C/D layout: VGPR r -> M = 16*mt + r + 8*half, N = 16*nt + l15.
#pragma unroll
  for (int r = 0; r < 8; ++r) {
    int o  = mt * 16 + r + half * 8;
    int b0 = nt0 * 16 + l15;
    int b1 = b0 + 16;
    if (mode == 0) {
      outp[((size_t)i * KOUT + o) * NB + b0] = acc0[r];
      outp[((size_t)i * KOUT + o) * NB + b1] = acc1[r];
    } else {
      float bv = bias[o];
      float g0 = Gs[((size_t)i * KOUT + o) * NB + b0];
      float g1 = Gs[((size_t)i * KOUT + o) * NB + b1];
      outp[((size_t)b0 * 24 + i) * (KOUT * 6) + o * 6 + d] = acc0[r] + g0 + bv;
      outp[((size_t)b1 * 24 + i) * (KOUT * 6) + o * 6 + d] = acc1[r] + g1 + bv;
    }
  }
}

// ---------------------------------------------------------------------------
extern "C" void kernel_launch(void* const* d_in, const int* in_sizes, int n_in,
                              void* d_out, int out_size, void* d_ws, size_t ws_size,
                              hipStream_t stream) {
  const float* x    = (const float*)d_in[0];     // (64, 24, 384)
  const float* w    = (const float*)d_in[1];     // (64, 64, 24, 2)
  const float* bias = (const float*)d_in[2];     // (64,)
  const int*   sp   = (const int*)d_in[3];       // (24, 24)
  // d_in[4] = co_orbit: structure (d != c) folded analytically into Wd/W1 split.
  float* out = (float*)d_out;

  float* Xm = (float*)d_ws;                      // 6*24*64*64 = 589824 floats
  float* Xs = Xm + 6 * NSP * NB * KIN;           // 98304
  float* Wd = Xs + NSP * NB * KIN;               // 98304
  float* W1 = Wd + NSP * KOUT * KIN;             // 98304
  float* Gs = W1 + NSP * KOUT * KIN;             // 98304
  (void)in_sizes; (void)n_in; (void)out_size; (void)ws_size;

  prep_kernel<<<dim3(384, 2), 256, 0, stream>>>(x, w, Xm, Xs, Wd, W1);
  // Pass 1: Gs[i][o][b] = sum_{j,k} W1[sp[i,j]][o][k] * S[j][b][k]
  gemm_kernel<<<dim3(24, 1), 256, LDS_TOTAL_BYTES, stream>>>(
      W1, Xs, sp, nullptr, nullptr, Gs, 0);
  // Pass 2: out = Wd-GEMM + Gs + bias, written in final (b, i, o*6+d) layout
  gemm_kernel<<<dim3(24, 6), 256, LDS_TOTAL_BYTES, stream>>>(
      Wd, Xm, sp, Gs, bias, out, 1);
}